// NPNetVLoss_70111046140309
// MI455X (gfx1250) — compile-verified
//
#include <hip/hip_runtime.h>
#include <math.h>

// ---------------------------------------------------------------------------
// NPNetVLoss for MI455X (gfx1250, wave32).
// Memory-bound streaming reduce: 268 MB traffic -> ~11.5 us roofline @23.3TB/s.
// Rank-4 DCT projection done as 8x V_WMMA_F32_16X16X4_F32 per 16x32 tile.
// Operand order chosen so the coefficient matrix comes out TRANSPOSED:
//   A (16x4)  = first-4 DCT-II rows (constant registers, rows 4..15 zero)
//   B (4x16)  = streamed e = x_star - x_target slice (pixel = lane)
//   C (16x16) = c^T : VGPR v holds coefficient v, lane holds pixel
// so the temporal-diff MSE = per-pixel quadratic form c^T Q c needs NO
// cross-lane shuffles at all (lanes 16..31 see zero rows and add 0).
// ---------------------------------------------------------------------------

typedef __attribute__((ext_vector_type(2))) float v2f;
typedef __attribute__((ext_vector_type(8))) float v8f;

#define TDIM      32
#define HW        16384          // 128*128
#define PLANE     16384          // element stride between consecutive t
#define BCSTRIDE  524288         // 32*16384, stride between (b,c) planes
#define NTILES    65536          // (4*16*128*128)/16 pixels per tile
#define NBLOCKS   2048
#define EPS2      1e-12f         // charbonnier eps^2  (eps = 1e-6)
#define PI_F      3.14159265358979323846f

__global__ __launch_bounds__(256) void npnetv_loss_main(
    const float* __restrict__ xs,     // x_star
    const float* __restrict__ xt,     // x_target
    double* __restrict__ ws)          // per-block partials: [2*blockIdx]
{
    __shared__ float sD[TDIM * 4];    // orthonormal DCT rows, t-major [t][n]
    __shared__ float sQ[10];          // upper triangle of Q = dD^T dD
    __shared__ double sCh[8], sQf[8]; // per-wave partials

    const int tid = threadIdx.x;

    // --- build DCT table D[n][t] (first 4 ortho DCT-II rows) in LDS --------
    if (tid < TDIM * 4) {
        int t = tid >> 2, n = tid & 3;
        float v = 0.25f * cosf(PI_F * (2.0f * (float)t + 1.0f) * (float)n / 64.0f);
        if (n == 0) v *= 0.70710678118654752f;   // DC row ortho scaling
        sD[t * 4 + n] = v;
    }
    __syncthreads();

    // --- Q[n][m] = sum_s (D[n][s+1]-D[n][s]) * (D[m][s+1]-D[m][s]) ----------
    if (tid < 10) {
        const int pn[10] = {0,1,2,3,0,0,0,1,1,2};
        const int pm[10] = {0,1,2,3,1,2,3,2,3,3};
        int n = pn[tid], m = pm[tid];
        float q = 0.0f;
        for (int s = 0; s < TDIM - 1; ++s) {
            float dn = sD[(s + 1) * 4 + n] - sD[s * 4 + n];
            float dm = sD[(s + 1) * 4 + m] - sD[s * 4 + m];
            q = fmaf(dn, dm, q);
        }
        sQ[tid] = q;
    }
    __syncthreads();

    const int lane = tid & 31;
    const int col  = lane & 15;   // A's M row (coeff) == B's N column (pixel)
    const int h    = lane >> 4;   // lane half selects K sub-pair

    // --- per-lane constant A-matrix registers for the 8 K-chunks ------------
    // A (16x4): lane m = row; VGPR0 -> K=0 (lanes 0-15) / K=2 (lanes 16-31),
    //           VGPR1 -> K=1 / K=3.  Rows m>=4 are zero.
    float a0r[8], a1r[8];
#pragma unroll
    for (int c = 0; c < 8; ++c) {
        int t0 = 4 * c + 2 * h;
        a0r[c] = (col < 4) ? sD[t0 * 4 + col]       : 0.0f;
        a1r[c] = (col < 4) ? sD[(t0 + 1) * 4 + col] : 0.0f;
    }
    const float q00 = sQ[0], q11 = sQ[1], q22 = sQ[2], q33 = sQ[3];
    const float q01 = sQ[4], q02 = sQ[5], q03 = sQ[6];
    const float q12 = sQ[7], q13 = sQ[8], q23 = sQ[9];

    float chSum = 0.0f;   // charbonnier partial (all lanes)
    float qfSum = 0.0f;   // quadratic-form partial (lanes 0-15 carry pixels)

    const int waveId = (blockIdx.x * blockDim.x + tid) >> 5;
    const int nWaves = (gridDim.x * blockDim.x) >> 5;

    for (int tile = waveId; tile < NTILES; tile += nWaves) {
        const int pix0 = tile << 4;                        // 16 px per tile
        const size_t base = (size_t)(pix0 >> 14) * (size_t)BCSTRIDE
                          + (size_t)(pix0 & (HW - 1)) + (size_t)col;

        v8f acc = {0.f, 0.f, 0.f, 0.f, 0.f, 0.f, 0.f, 0.f};

#pragma unroll
        for (int c = 0; c < 8; ++c) {
            const int t0 = 4 * c + 2 * h;                  // this half's t pair
            const float* pa = xs + base + (size_t)t0 * PLANE;
            const float* pb = xt + base + (size_t)t0 * PLANE;
            // coalesced: each load = 2 x 64B rows (one per lane-half);
            // adjacent tiles -> adjacent waves, so L2 merges 128B lines.
            float e0 = pa[0]     - pb[0];
            float e1 = pa[PLANE] - pb[PLANE];

            // raw v_sqrt_f32: args >= 1e-12, far above denormal range,
            // so skip libm's scale-check scaffolding.
            chSum += __builtin_amdgcn_sqrtf(fmaf(e0, e0, EPS2))
                   + __builtin_amdgcn_sqrtf(fmaf(e1, e1, EPS2));

            v2f av; av.x = a0r[c]; av.y = a1r[c];          // A: DCT rows (const)
            v2f bv; bv.x = e0;     bv.y = e1;              // B: 4x16 slice of e
            // C += A x B : coefficient n in VGPR n, pixel in lane.
            acc = __builtin_amdgcn_wmma_f32_16x16x4_f32(
                false, av, false, bv, (short)0, acc, false, false);
        }

        // --- per-pixel quadratic form c^T Q c, no shuffles ------------------
        // lanes 0-15: acc[n] = coeff n of pixel `lane`; lanes 16-31: zeros.
        {
            float s0 = acc[0], s1 = acc[1], s2 = acc[2], s3 = acc[3];
            qfSum += q00*s0*s0 + q11*s1*s1 + q22*s2*s2 + q33*s3*s3
                   + 2.0f*(q01*s0*s1 + q02*s0*s2 + q03*s0*s3
                         + q12*s1*s2 + q13*s1*s3 + q23*s2*s3);
        }
    }

    // --- wave32 reduce, then per-block partial in double --------------------
    for (int off = 16; off; off >>= 1) {
        chSum += __shfl_xor(chSum, off, 32);
        qfSum += __shfl_xor(qfSum, off, 32);
    }
    if (lane == 0) { sCh[tid >> 5] = (double)chSum; sQf[tid >> 5] = (double)qfSum; }
    __syncthreads();
    if (tid == 0) {
        double c = 0.0, q = 0.0;
        for (int i = 0; i < 8; ++i) { c += sCh[i]; q += sQf[i]; }
        ws[2 * blockIdx.x]     = c;
        ws[2 * blockIdx.x + 1] = q;
    }
}

// ---------------------------------------------------------------------------
// Deterministic final reduce over the 2048 block partials + scalar math.
// ---------------------------------------------------------------------------
__global__ __launch_bounds__(256) void npnetv_loss_finalize(
    const double* __restrict__ ws, float* __restrict__ out)
{
    __shared__ double sc[256], sq[256];
    double c = 0.0, q = 0.0;
    for (int i = threadIdx.x; i < NBLOCKS; i += 256) {
        c += ws[2 * i];
        q += ws[2 * i + 1];
    }
    sc[threadIdx.x] = c; sq[threadIdx.x] = q;
    __syncthreads();
    for (int s = 128; s; s >>= 1) {
        if (threadIdx.x < s) {
            sc[threadIdx.x] += sc[threadIdx.x + s];
            sq[threadIdx.x] += sq[threadIdx.x + s];
        }
        __syncthreads();
    }
    if (threadIdx.x == 0) {
        double Lmain = sc[0] / 33554432.0;   // 4*16*32*128*128
        double Ltemp = sq[0] / 32505856.0;   // 4*16*31*128*128
        out[0] = (float)(Lmain + 0.1 * Ltemp);
        out[1] = (float)Lmain;
        out[2] = (float)Ltemp;
    }
}

extern "C" void kernel_launch(void* const* d_in, const int* in_sizes, int n_in,
                              void* d_out, int out_size, void* d_ws, size_t ws_size,
                              hipStream_t stream)
{
    const float* xs = (const float*)d_in[0];   // x_star_T  (f32)
    const float* xt = (const float*)d_in[1];   // x_target_T (f32)
    float* out = (float*)d_out;                // 3 scalars: total, L_main, L_temp
    double* ws = (double*)d_ws;                // 2048 * 2 doubles = 32 KB

    npnetv_loss_main<<<NBLOCKS, 256, 0, stream>>>(xs, xt, ws);
    npnetv_loss_finalize<<<1, 256, 0, stream>>>(ws, out);
}